// JointLoss_3882650436521
// MI455X (gfx1250) — compile-verified
//
#include <hip/hip_runtime.h>

// JointLoss for MI455X (gfx1250, wave32).
// Roofline: similarity streams = 524 MB -> ~22.5us @ 23.3 TB/s.
// GEMM work = 33.6 GFLOP fp32 via V_WMMA_F32_16X16X4_F32.
//
// v3: double-buffered B-fragment preload (4 chunks x 8 b64 loads) keeps
// load/WMMA overlap while cutting peak VGPR pressure under 256 (v2 spilled
// into the VGPR-MSB region, costing s_set_vgpr_msb churn + occupancy).
// Similarity values for the tile are preloaded before the WMMA loop so the
// epilogue never stalls on HBM.

typedef float v2f __attribute__((ext_vector_type(2)));
typedef float v8f __attribute__((ext_vector_type(8)));

#define B_ROWS 16384
#define C_COLS 4000
#define D_DIM  128
#define NTILES (C_COLS / 16)   // 250 column tiles of 16
#define LDS_STRIDE 132         // 128 + 4 pad: bank = (4*lm + 2*hi) -> conflict-free

// ---------------------------------------------------------------- zero ws
__global__ void zero_ws_kernel(float* ws) {
    if (threadIdx.x < 3) ws[threadIdx.x] = 0.0f;
}

// ---------------------------------------------------- loss_pos (exact term)
__global__ __launch_bounds__(256) void loss_pos_kernel(
    const float* __restrict__ F, const float* __restrict__ A,
    const int* __restrict__ labels, float* __restrict__ ws) {
    const int wave = threadIdx.x >> 5;
    const int lane = threadIdx.x & 31;
    const int row  = blockIdx.x * 8 + wave;
    const int lbl  = labels[row];
    const float4 f = ((const float4*)(F + (size_t)row * D_DIM))[lane];
    const float4 a = ((const float4*)(A + (size_t)lbl * D_DIM))[lane];
    const float dx = f.x - a.x, dy = f.y - a.y, dz = f.z - a.z, dw = f.w - a.w;
    float s = dx * dx + dy * dy + dz * dz + dw * dw;
    #pragma unroll
    for (int m = 1; m < 32; m <<= 1) s += __shfl_xor(s, m, 32);
    if (lane == 0) atomicAdd(ws + 0, s);
}

// ------------------------------------------------------------- main kernel
// Workgroup = 16 rows of B; 8 waves each own every-8th 16-col tile.
// Per tile: 32 K-slices * 2 GEMMs = 64 v_wmma_f32_16x16x4_f32.
__global__ __launch_bounds__(256) void joint_tile_kernel(
    const float* __restrict__ F,  const float* __restrict__ A,
    const int*   __restrict__ labels,
    const float* __restrict__ S,  const float* __restrict__ Ft,
    const float* __restrict__ St, float* __restrict__ ws) {

    __shared__ float sF [16 * LDS_STRIDE];   // F  stripe, padded rows
    __shared__ float sFt[16 * LDS_STRIDE];   // Ft stripe, padded rows
    __shared__ float sFn2[16], sFtn2[16];
    __shared__ int   sLab[16];
    __shared__ float sHS[16], sCS[16], sHT[16], sCT[16];

    const int b0   = blockIdx.x * 16;
    const int t    = threadIdx.x;
    const int wave = t >> 5;
    const int lane = t & 31;
    const int lm   = lane & 15;     // N / column within tile
    const int hi   = lane >> 4;     // K-half selector for A/B fragments

    if (t < 16) {
        sHS[t] = 0.f; sCS[t] = 0.f; sHT[t] = 0.f; sCT[t] = 0.f;
        sLab[t] = labels[b0 + t];
    }

    // Stage the 16x128 F and Ft stripes into padded LDS (float4 copies).
    #pragma unroll
    for (int it = 0; it < 2; ++it) {
        const int idx = t + it * 256;        // 0..511 -> (row, float4-col)
        const int r  = idx >> 5;
        const int j4 = idx & 31;
        const float4 v = ((const float4*)(F  + (size_t)(b0 + r) * D_DIM))[j4];
        const float4 w = ((const float4*)(Ft + (size_t)(b0 + r) * D_DIM))[j4];
        *(float4*)&sF [r * LDS_STRIDE + j4 * 4] = v;
        *(float4*)&sFt[r * LDS_STRIDE + j4 * 4] = w;
    }
    __syncthreads();

    // Row norms from LDS (16 threads per row).
    {
        const int row = t >> 4, j0 = (t & 15) * 8;
        const float* fp = sF  + row * LDS_STRIDE + j0;
        const float* gp = sFt + row * LDS_STRIDE + j0;
        float sa = 0.f, sb = 0.f;
        #pragma unroll
        for (int j = 0; j < 8; ++j) { sa += fp[j] * fp[j]; sb += gp[j] * gp[j]; }
        if ((t & 15) == 0) { sFn2[row] = 0.f; sFtn2[row] = 0.f; }
        __syncthreads();
        atomicAdd(&sFn2[row],  sa);
        atomicAdd(&sFtn2[row], sb);
    }
    __syncthreads();

    float hS[8] = {0,0,0,0,0,0,0,0}, cS[8] = {0,0,0,0,0,0,0,0};
    float hT[8] = {0,0,0,0,0,0,0,0}, cT[8] = {0,0,0,0,0,0,0,0};

    // Per-lane LDS base for A-matrix fragments:
    // lanes 0-15: M=lane, K=4kb+{0,1}; lanes 16-31: M=lane-16, K=4kb+{2,3}.
    const float* fl = sF  + lm * LDS_STRIDE + 2 * hi;
    const float* gl = sFt + lm * LDS_STRIDE + 2 * hi;

    for (int tile = wave; tile < NTILES; tile += 8) {
        const int c0 = tile * 16;
        const int c  = c0 + lm;

        // Prefetch this wave's NEXT tile's similarity rows (global_prefetch_b8).
        {
            const int nc = (c0 + 128 < C_COLS) ? (c0 + 128) : c0;  // wave-uniform
            __builtin_prefetch(&S [(size_t)(b0 + lm) * C_COLS + nc]);
            __builtin_prefetch(&St[(size_t)(b0 + lm) * C_COLS + nc]);
        }

        // Preload this tile's similarity values (overlaps with WMMA below).
        float svS[8], svT[8];
        #pragma unroll
        for (int v = 0; v < 8; ++v) {
            const size_t bg = (size_t)(b0 + v + 8 * hi) * C_COLS + c;
            svS[v] = S [bg];
            svT[v] = St[bg];
        }

        // B fragments: column c0+lm of A^T == row (c0+lm) of agents.
        // Double-buffered: 4 chunks of 8 b64 loads, next chunk issued while
        // current chunk's 16 WMMAs run.
        const float* arow = A + (size_t)c * D_DIM + 2 * hi;
        v2f bb[2][8];
        #pragma unroll
        for (int j = 0; j < 8; ++j) bb[0][j] = *(const v2f*)(arow + 4 * j);

        v8f accF = {}; v8f accG = {};
        float colsq = 0.f;
        #pragma unroll
        for (int ch = 0; ch < 4; ++ch) {
            if (ch < 3) {
                #pragma unroll
                for (int j = 0; j < 8; ++j)
                    bb[(ch + 1) & 1][j] = *(const v2f*)(arow + 32 * (ch + 1) + 4 * j);
            }
            #pragma unroll
            for (int j = 0; j < 8; ++j) {
                const int kb = ch * 8 + j;
                const v2f b  = bb[ch & 1][j];
                colsq += b.x * b.x + b.y * b.y;       // inline ||a_c||^2 partial
                const v2f fa = *(const v2f*)(fl + 4 * kb);
                const v2f ga = *(const v2f*)(gl + 4 * kb);
                accF = __builtin_amdgcn_wmma_f32_16x16x4_f32(
                    false, fa, false, b, (short)0, accF, false, false);
                accG = __builtin_amdgcn_wmma_f32_16x16x4_f32(
                    false, ga, false, b, (short)0, accG, false, false);
            }
        }
        const float an2 = colsq + __shfl_xor(colsq, 16, 32);  // full-K col norm

        // C/D layout: VGPR v -> row (v + 8*hi), col = lm.
        #pragma unroll
        for (int v = 0; v < 8; ++v) {
            const int m = v + 8 * hi;
            const float distF = sFn2[m]  + an2 - 2.f * accF[v];
            if ((svS[v] > 0.5f) && (c != sLab[m])) {
                const float h = 1.f - distF;
                hS[v] += (h > 0.f) ? h : 0.f;
                cS[v] += 1.f;
            }
            const float distT = sFtn2[m] + an2 - 2.f * accG[v];
            if (svT[v] > 0.5f) {
                const float h = 1.f - distT;
                hT[v] += (h > 0.f) ? h : 0.f;
                cT[v] += 1.f;
            }
        }
    }

    // Reduce over the 16 columns held by lanes sharing the same hi.
    #pragma unroll
    for (int v = 0; v < 8; ++v) {
        #pragma unroll
        for (int m = 1; m < 16; m <<= 1) {
            hS[v] += __shfl_xor(hS[v], m, 32);
            cS[v] += __shfl_xor(cS[v], m, 32);
            hT[v] += __shfl_xor(hT[v], m, 32);
            cT[v] += __shfl_xor(cT[v], m, 32);
        }
    }
    if (lm == 0) {
        #pragma unroll
        for (int v = 0; v < 8; ++v) {
            const int m = v + 8 * hi;
            atomicAdd(&sHS[m], hS[v]); atomicAdd(&sCS[m], cS[v]);
            atomicAdd(&sHT[m], hT[v]); atomicAdd(&sCT[m], cT[v]);
        }
    }
    __syncthreads();

    if (t < 16) {
        const float cs = sCS[t], ct2 = sCT[t];
        const float part  = ((cs  > 0.f) ? sHS[t] / cs  : 0.f)
                          + ((ct2 > 0.f) ? sHT[t] / ct2 : 0.f);
        const float valid = ((cs > 0.f) ? 1.f : 0.f) + ((ct2 > 0.f) ? 1.f : 0.f);
        atomicAdd(ws + 1, part);
        atomicAdd(ws + 2, valid);
    }
}

// ---------------------------------------------------------------- finalize
__global__ void finalize_kernel(const float* __restrict__ ws,
                                float* __restrict__ out) {
    out[0] = (ws[0] + ws[1]) / (16384.0f + ws[2]);
}

// ----------------------------------------------------------------- launch
extern "C" void kernel_launch(void* const* d_in, const int* in_sizes, int n_in,
                              void* d_out, int out_size, void* d_ws, size_t ws_size,
                              hipStream_t stream) {
    const float* F      = (const float*)d_in[0];   // features        (B,D)
    const float* A      = (const float*)d_in[1];   // agents          (C,D)
    const int*   labels = (const int*)  d_in[2];   // labels          (B,)
    const float* S      = (const float*)d_in[3];   // similarity      (B,C)
    const float* Ft     = (const float*)d_in[4];   // features_target (B,D)
    const float* St     = (const float*)d_in[5];   // similarity_tgt  (B,C)
    float* out = (float*)d_out;
    float* ws  = (float*)d_ws;   // [0]=pos_sum [1]=neg_sum [2]=valid_cnt

    zero_ws_kernel<<<1, 64, 0, stream>>>(ws);
    loss_pos_kernel<<<B_ROWS / 8, 256, 0, stream>>>(F, A, labels, ws);
    joint_tile_kernel<<<B_ROWS / 16, 256, 0, stream>>>(F, A, labels, S, Ft, St, ws);
    finalize_kernel<<<1, 1, 0, stream>>>(ws, out);
}